// RadialBasisFunction_40518721471189
// MI455X (gfx1250) — compile-verified
//
#include <hip/hip_runtime.h>
#include <hip/hip_bf16.h>

// ---------------------------------------------------------------------------
// out[b,o] = ||x_b - w_o||^2 = |x_b|^2 - 2 x_b.w_o + |w_o|^2
// Cross term via V_WMMA_F32_16X16X4_F32 (full f32 precision, memory-bound
// kernel: ~161MB traffic @ 23.3TB/s ~= 7us; compute need ~2.5TF << f32 WMMA peak).
// |x|^2 and |w|^2 accumulated in-loop from the same fragments, fused epilogue.
// ---------------------------------------------------------------------------

typedef float v2f  __attribute__((ext_vector_type(2)));
typedef float v8f  __attribute__((ext_vector_type(8)));

#define BDIM   32768
#define INDIM  256
#define OUTDIM 1024

#define M_TILE 128   // 8 waves x 16 rows
#define N_TILE 64    // 4 x 16-col WMMA strips per wave
#define K_CHUNK 64

#define XS_STRIDE (K_CHUNK + 4)   // padded LDS strides (floats) -> no bank conflicts
#define WS_STRIDE (N_TILE + 4)

__global__ __launch_bounds__(256) void
rbf_dist2_wmma_f32(const float* __restrict__ x,
                   const float* __restrict__ w,
                   float* __restrict__ out)
{
    __shared__ float xs[M_TILE * XS_STRIDE];   // 128 x 68 floats = 34816 B
    __shared__ float ws[K_CHUNK * WS_STRIDE];  //  64 x 68 floats = 17408 B

    const int t    = threadIdx.x;
    const int wave = t >> 5;        // 0..7
    const int lane = t & 31;
    const int hi   = lane >> 4;     // half-wave select
    const int mrow = lane & 15;

    const int n0 = blockIdx.x * N_TILE;   // 16 tiles over OUT
    const int m0 = blockIdx.y * M_TILE;   // 256 tiles over B

    // staging decomposition: 16 threads per row, float4 each
    const int cg = t & 15;          // column group (16B granules)
    const int rg = t >> 4;          // 0..15

    v8f acc[4] = {};
    float xsq_part = 0.0f;
    float wsq_part[4] = {0.0f, 0.0f, 0.0f, 0.0f};

    for (int kc = 0; kc < INDIM; kc += K_CHUNK) {
        // ---- stage x tile: rows [m0, m0+128), cols [kc, kc+64) ----
        #pragma unroll
        for (int r = 0; r < 8; ++r) {
            const int row = rg + r * 16;
            const float4 v = *(const float4*)(x + (size_t)(m0 + row) * INDIM + kc + cg * 4);
            *(float4*)(&xs[row * XS_STRIDE + cg * 4]) = v;
        }
        // ---- stage w tile: K rows [kc, kc+64), cols [n0, n0+64) ----
        #pragma unroll
        for (int r = 0; r < 4; ++r) {
            const int krow = rg + r * 16;
            const float4 v = *(const float4*)(w + (size_t)(kc + krow) * OUTDIM + n0 + cg * 4);
            *(float4*)(&ws[krow * WS_STRIDE + cg * 4]) = v;
        }
        // prefetch next x chunk into cache hierarchy (global_prefetch_b8)
        if (kc + K_CHUNK < INDIM) {
            __builtin_prefetch(x + (size_t)(m0 + rg) * INDIM + kc + K_CHUNK + cg * 4, 0, 1);
        }
        __syncthreads();

        // ---- 16 x (k+=4) WMMA steps over this chunk ----
        #pragma unroll 4
        for (int kk = 0; kk < K_CHUNK; kk += 4) {
            // A fragment 16x4: lanes 0-15 hold K=kk,kk+1; lanes 16-31 hold K=kk+2,kk+3
            const v2f a = *(const v2f*)(&xs[(wave * 16 + mrow) * XS_STRIDE + kk + 2 * hi]);
            xsq_part += a.x * a.x + a.y * a.y;

            #pragma unroll
            for (int j = 0; j < 4; ++j) {
                // B fragment 4x16: vgpr0 rows K=kk+hi, vgpr1 rows K=kk+2+hi
                v2f b;
                b.x = ws[(kk + hi)     * WS_STRIDE + j * 16 + mrow];
                b.y = ws[(kk + 2 + hi) * WS_STRIDE + j * 16 + mrow];
                wsq_part[j] += b.x * b.x + b.y * b.y;

                acc[j] = __builtin_amdgcn_wmma_f32_16x16x4_f32(
                    /*neg_a=*/false, a, /*neg_b=*/false, b,
                    /*c_mod=*/(short)0, acc[j],
                    /*reuse_a=*/false, /*reuse_b=*/false);
            }
        }
        __syncthreads();
    }

    // ---- combine half-wave partials (each half saw half of K) ----
    const float xsq_full = xsq_part + __shfl_xor(xsq_part, 16, 32);
    float wsq_full[4];
    #pragma unroll
    for (int j = 0; j < 4; ++j)
        wsq_full[j] = wsq_part[j] + __shfl_xor(wsq_part[j], 16, 32);

    // ---- fused epilogue: out = xsq - 2*cross + wsq ----
    // C/D layout: vgpr e -> row M = e + 8*hi, col N = mrow
    #pragma unroll
    for (int e = 0; e < 8; ++e) {
        const int rowInTile = wave * 16 + e + 8 * hi;
        // row sums live in lanes whose (lane&15) == rowIdx within the wave strip
        const float xsq_e = __shfl(xsq_full, e + 8 * hi, 32);
        const size_t gRow = (size_t)(m0 + rowInTile);
        #pragma unroll
        for (int j = 0; j < 4; ++j) {
            const int gCol = n0 + j * 16 + mrow;
            out[gRow * OUTDIM + gCol] = xsq_e - 2.0f * acc[j][e] + wsq_full[j];
        }
    }
}

extern "C" void kernel_launch(void* const* d_in, const int* in_sizes, int n_in,
                              void* d_out, int out_size, void* d_ws, size_t ws_size,
                              hipStream_t stream) {
    (void)in_sizes; (void)n_in; (void)d_ws; (void)ws_size; (void)out_size;
    const float* x = (const float*)d_in[0];   // [32768, 256]
    const float* w = (const float*)d_in[1];   // [1, 256, 1024] -> [256, 1024]
    float* out = (float*)d_out;               // [32768, 1024]

    dim3 grid(OUTDIM / N_TILE, BDIM / M_TILE);  // (16, 256)
    dim3 block(256);
    rbf_dist2_wmma_f32<<<grid, block, 0, stream>>>(x, w, out);
}